// CoverageMechanism_2173253452368
// MI455X (gfx1250) — compile-verified
//
#include <hip/hip_runtime.h>
#include <hip/hip_bf16.h>
#include <math.h>

// Problem constants (from reference setup_inputs)
constexpr int B = 8;
constexpr int T = 100;
constexpr int S = 400;
constexpr int H = 256;
constexpr float MAX_LOGIT = 30.0f;
constexpr float COV_WEIGHT = 1.0f;

typedef __attribute__((ext_vector_type(2))) float v2f;
typedef __attribute__((ext_vector_type(8))) float v8f;
typedef __attribute__((ext_vector_type(4))) int v4i;

// ---------------------------------------------------------------------------
// gfx1250 feature shims (guarded so compile never breaks)
// ---------------------------------------------------------------------------
#if __has_builtin(__builtin_amdgcn_tanhf)
#define FAST_TANH(x) __builtin_amdgcn_tanhf(x)
#elif __has_builtin(__builtin_amdgcn_tanh_f32)
#define FAST_TANH(x) __builtin_amdgcn_tanh_f32(x)
#else
#define FAST_TANH(x) tanhf(x)
#endif

#if __has_builtin(__builtin_amdgcn_global_load_async_to_lds_b128)
#define HAS_ASYNC_LDS 1
#else
#define HAS_ASYNC_LDS 0
#endif

// Builtin prototype (from hipcc diagnostic): pointers to 4xi32 vectors,
// global src in addrspace(1), LDS dst in addrspace(3), imm offset, imm cpol.
typedef __attribute__((address_space(1))) v4i gv4i_t;
typedef __attribute__((address_space(3))) v4i lv4i_t;

__device__ __forceinline__ void async_copy_b128(const void* gsrc, void* ldst) {
#if HAS_ASYNC_LDS
  __builtin_amdgcn_global_load_async_to_lds_b128(
      (gv4i_t*)(uintptr_t)gsrc,
      (lv4i_t*)(uint32_t)(uintptr_t)ldst,  // low 32 bits of flat addr = LDS offset
      /*offset=*/0, /*cpol=*/0);
#else
  (void)gsrc; (void)ldst;
#endif
}
__device__ __forceinline__ void wait_async_le2() {
#if __has_builtin(__builtin_amdgcn_s_wait_asynccnt)
  __builtin_amdgcn_s_wait_asynccnt(2);
#else
  asm volatile("s_wait_asynccnt 0x2" ::: "memory");
#endif
  asm volatile("" ::: "memory");  // keep LDS reads below the wait
}
__device__ __forceinline__ void wait_async_le0() {
#if __has_builtin(__builtin_amdgcn_s_wait_asynccnt)
  __builtin_amdgcn_s_wait_asynccnt(0);
#else
  asm volatile("s_wait_asynccnt 0x0" ::: "memory");
#endif
  asm volatile("" ::: "memory");
}

// ---------------------------------------------------------------------------
// GEMM: Y[M,N] = X[M,K] @ W[K,N] (+ bias[N]), f32, via V_WMMA_F32_16X16X4_F32
// One wave (32 threads) per 16x16 output tile. M,N,K all multiples of 16/4.
// A 16x4 layout: lanes 0-15 -> (K=k0,k0+1), lanes 16-31 -> (K=k0+2,k0+3),
// row = lane&15.  B 4x16 mirrored: col = lane&15, same K split.
// C/D: VGPR r -> M = r (lanes 0-15) / M = r+8 (lanes 16-31), N = lane&15.
// ---------------------------------------------------------------------------
__global__ void gemm_wmma_f32(const float* __restrict__ X,
                              const float* __restrict__ W,
                              const float* __restrict__ bias,
                              float* __restrict__ Y,
                              int M, int N, int K) {
  const int mTile = blockIdx.x;
  const int nTile = blockIdx.y;
  const int lane  = threadIdx.x;           // 0..31, one wave
  const int hi    = lane >> 4;             // 0 or 1 (K half)
  const int l15   = lane & 15;
  const int row   = mTile * 16 + l15;      // A row for this lane
  const int col   = nTile * 16 + l15;      // B col / C col for this lane

  v8f acc = {};
#pragma unroll 4
  for (int k0 = 0; k0 < K; k0 += 4) {
    const int ka = k0 + hi * 2;
    v2f a, b;
    a.x = X[(size_t)row * K + ka];
    a.y = X[(size_t)row * K + ka + 1];
    b.x = W[(size_t)ka * N + col];
    b.y = W[(size_t)(ka + 1) * N + col];
    acc = __builtin_amdgcn_wmma_f32_16x16x4_f32(
        /*neg_a=*/false, a, /*neg_b=*/false, b,
        /*c_mod=*/(short)0, acc, /*reuse_a=*/false, /*reuse_b=*/false);
  }

  const int orow = mTile * 16 + hi * 8;
  const float bv = bias ? bias[col] : 0.0f;
#pragma unroll
  for (int r = 0; r < 8; ++r) {
    Y[(size_t)(orow + r) * N + col] = acc[r] + bv;
  }
}

// ---------------------------------------------------------------------------
// Reductions (wave32)
// ---------------------------------------------------------------------------
__device__ __forceinline__ float waveSum(float v) {
#pragma unroll
  for (int o = 16; o > 0; o >>= 1) v += __shfl_xor(v, o, 32);
  return v;
}
__device__ __forceinline__ float waveMax(float v) {
#pragma unroll
  for (int o = 16; o > 0; o >>= 1) v = fmaxf(v, __shfl_xor(v, o, 32));
  return v;
}
__device__ float blockSum(float v, float* tmp) {  // 256 threads = 8 waves
  const int lane = threadIdx.x & 31, w = threadIdx.x >> 5;
  v = waveSum(v);
  __syncthreads();
  if (lane == 0) tmp[w] = v;
  __syncthreads();
  if (w == 0) {
    float x = (lane < 8) ? tmp[lane] : 0.0f;
    x = waveSum(x);
    if (lane == 0) tmp[0] = x;
  }
  __syncthreads();
  return tmp[0];
}
__device__ float blockMax(float v, float* tmp) {
  const int lane = threadIdx.x & 31, w = threadIdx.x >> 5;
  v = waveMax(v);
  __syncthreads();
  if (lane == 0) tmp[w] = v;
  __syncthreads();
  if (w == 0) {
    float x = (lane < 8) ? tmp[lane] : -__builtin_inff();
    x = waveMax(x);
    if (lane == 0) tmp[0] = x;
  }
  __syncthreads();
  return tmp[0];
}

// ---------------------------------------------------------------------------
// Scores + softmax. One block per (b,t). 256 threads = 8 waves.
// Each lane owns a fixed 8-wide H chunk (h = (lane&31)*8 .. +7) so v, w_c and
// Ws[b,t,:] stay in registers across all 400 s values; wave w handles
// s = w, w+8, ... (exactly 50 rows each).
// Wh rows are double-buffered into LDS with GLOBAL_LOAD_ASYNC_TO_LDS_B128:
// ASYNCcnt is wave-private and async loads complete in order, so after
// issuing row i+1 (2 b128s/lane), s_wait_asynccnt<=2 guarantees row i landed
// — no barrier needed since only the issuing wave consumes its buffers.
// ---------------------------------------------------------------------------
__global__ void scores_softmax(const float* __restrict__ Wh,   // [B,S,H] (bias folded)
                               const float* __restrict__ Ws,   // [B,T,H]
                               const float* __restrict__ v,
                               const float* __restrict__ wc,
                               const unsigned char* __restrict__ mask,  // [B,S]
                               const float* __restrict__ cov,  // [B,T,S] or null
                               float* __restrict__ A) {        // [B,T,S]
  const int bt = blockIdx.x;
  const int b  = bt / T;
  __shared__ float sE[S];
  __shared__ float tmp[8];
#if HAS_ASYNC_LDS
  __shared__ __align__(16) float sWh[8][2][H];  // 16 KB: per-wave double buffer
#endif

  const int tid  = threadIdx.x;
  const int lane = tid & 31;
  const int w    = tid >> 5;
  const int h0   = lane * 8;

  float vreg[8], wcreg[8], wst[8];
#pragma unroll
  for (int j = 0; j < 8; ++j) {
    vreg[j]  = v[h0 + j];
    wcreg[j] = wc[h0 + j];
    wst[j]   = Ws[(size_t)bt * H + h0 + j];
  }

  const float* WhB = Wh + (size_t)b * S * H;
  const unsigned char* mrow = mask + (size_t)b * S;
  const bool use_cov = (cov != nullptr);

#if HAS_ASYNC_LDS
  {  // prologue: fetch this wave's first row (s = w) into buffer 0
    const float* g = WhB + (size_t)w * H + h0;
    float* l = &sWh[w][0][h0];
    async_copy_b128(g, l);
    async_copy_b128(g + 4, l + 4);
  }
  int buf = 0;
#endif

  for (int s = w; s < S; s += 8) {
    const float c = use_cov ? cov[(size_t)bt * S + s] : 0.0f;
#if HAS_ASYNC_LDS
    const int snext = s + 8;
    if (snext < S) {
      const float* g = WhB + (size_t)snext * H + h0;
      float* l = &sWh[w][buf ^ 1][h0];
      async_copy_b128(g, l);
      async_copy_b128(g + 4, l + 4);
      wait_async_le2();   // the 2 oldest (current row) have landed
    } else {
      wait_async_le0();
    }
    const float* rowp = &sWh[w][buf][h0];
    buf ^= 1;
#else
    const float* rowp = WhB + (size_t)s * H + h0;
#endif
    float acc = 0.0f;
#pragma unroll
    for (int j = 0; j < 8; ++j) {
      const float x = rowp[j] + wst[j] + c * wcreg[j];
      acc = fmaf(vreg[j], FAST_TANH(x), acc);
    }
    acc = waveSum(acc);
    if (lane == 0) {
      acc = fminf(fmaxf(acc, -MAX_LOGIT), MAX_LOGIT);
      if (mrow[s]) acc = -__builtin_inff();
      sE[s] = acc;
    }
  }
  __syncthreads();

  // softmax over S in LDS
  float m = -__builtin_inff();
  for (int s = tid; s < S; s += 256) m = fmaxf(m, sE[s]);
  m = blockMax(m, tmp);

  float sum = 0.0f;
  for (int s = tid; s < S; s += 256) {
    const float e = __expf(sE[s] - m);
    sE[s] = e;
    sum += e;
  }
  sum = blockSum(sum, tmp);
  const float inv = 1.0f / sum;
  for (int s = tid; s < S; s += 256) A[(size_t)bt * S + s] = sE[s] * inv;
}

// ---------------------------------------------------------------------------
// cov_shifted[b,t,s] = sum_{t'<t} A_prelim[b,t',s].  Parallel over (b,s),
// serial over T (T=100). Thread 0 also zeroes the loss accumulator slot.
// ---------------------------------------------------------------------------
__global__ void cumsum_shift(const float* __restrict__ A,
                             float* __restrict__ cov,
                             float* __restrict__ loss_slot) {
  const int idx = blockIdx.x * blockDim.x + threadIdx.x;  // over B*S
  if (idx == 0) *loss_slot = 0.0f;
  if (idx >= B * S) return;
  const int b = idx / S, s = idx % S;
  float run = 0.0f;
  for (int t = 0; t < T; ++t) {
    const size_t off = ((size_t)(b * T + t)) * S + s;
    cov[off] = run;            // shifted: value before including row t
    run += A[off];
  }
}

// ---------------------------------------------------------------------------
// context[b,t,h] = sum_s A_final[b,t,s] * enc[b,s,h]; coverage loss partials.
// One block per (b,t), 256 threads; thread tid owns h = tid (coalesced enc
// reads across the block), A row cached in LDS.
// ---------------------------------------------------------------------------
__global__ void context_loss(const float* __restrict__ Afin,  // [B,T,S]
                             const float* __restrict__ cov,   // [B,T,S]
                             const float* __restrict__ enc,   // [B,S,H]
                             float* __restrict__ ctx,         // [B,T,H]
                             float* __restrict__ loss) {
  const int bt = blockIdx.x;
  const int b  = bt / T;
  __shared__ float sA[S];
  __shared__ float tmp[8];
  const int tid = threadIdx.x;

  float part = 0.0f;
  for (int s = tid; s < S; s += 256) {
    const float a = Afin[(size_t)bt * S + s];
    sA[s] = a;
    part += fminf(a, cov[(size_t)bt * S + s]);
  }
  __syncthreads();

  const float* eb = enc + (size_t)b * S * H;
  float acc = 0.0f;
  for (int s = 0; s < S; ++s) acc = fmaf(sA[s], eb[(size_t)s * H + tid], acc);
  ctx[(size_t)bt * H + tid] = acc;

  const float tot = blockSum(part, tmp);
  if (tid == 0) atomicAdd(loss, tot * (COV_WEIGHT / (float)(B * T)));
}

// ---------------------------------------------------------------------------
// Launch
// ---------------------------------------------------------------------------
extern "C" void kernel_launch(void* const* d_in, const int* in_sizes, int n_in,
                              void* d_out, int out_size, void* d_ws, size_t ws_size,
                              hipStream_t stream) {
  const float*         dec    = (const float*)d_in[0];          // [B,T,H]
  const float*         enc    = (const float*)d_in[1];          // [B,S,H]
  const unsigned char* mask   = (const unsigned char*)d_in[2];  // [B,S] bool
  const float*         W_h    = (const float*)d_in[3];          // [H,H]
  const float*         W_s    = (const float*)d_in[4];          // [H,H]
  const float*         w_c    = (const float*)d_in[5];          // [H]
  const float*         v      = (const float*)d_in[6];          // [H]
  const float*         b_attn = (const float*)d_in[7];          // [H]

  // Workspace layout (floats): Wh | Ws | A_prelim | cov_shifted  (~6.7 MB)
  float* ws   = (float*)d_ws;
  float* Wh   = ws;                         // B*S*H = 819200
  float* Wsp  = Wh  + (size_t)B * S * H;    // B*T*H = 204800
  float* Apre = Wsp + (size_t)B * T * H;    // B*T*S = 320000
  float* cov  = Apre + (size_t)B * T * S;   // B*T*S = 320000

  // Output layout: context [B,T,H] | A_final [B,T,S] | cov_loss [1]
  float* ctx  = (float*)d_out;
  float* Afin = ctx + (size_t)B * T * H;
  float* loss = Afin + (size_t)B * T * S;

  // 1) Wh = enc @ W_h + b_attn   (M=B*S=3200, N=K=H=256)
  gemm_wmma_f32<<<dim3((B * S) / 16, H / 16), dim3(32), 0, stream>>>(
      enc, W_h, b_attn, Wh, B * S, H, H);
  // 2) Ws = dec @ W_s            (M=B*T=800)
  gemm_wmma_f32<<<dim3((B * T) / 16, H / 16), dim3(32), 0, stream>>>(
      dec, W_s, nullptr, Wsp, B * T, H, H);
  // 3) preliminary scores + softmax
  scores_softmax<<<B * T, 256, 0, stream>>>(Wh, Wsp, v, w_c, mask, nullptr, Apre);
  // 4) shifted cumsum over T -> coverage; zero loss slot
  cumsum_shift<<<(B * S + 255) / 256, 256, 0, stream>>>(Apre, cov, loss);
  // 5) final scores (with coverage) + softmax
  scores_softmax<<<B * T, 256, 0, stream>>>(Wh, Wsp, v, w_c, mask, cov, Afin);
  // 6) context + coverage loss
  context_loss<<<B * T, 256, 0, stream>>>(Afin, cov, enc, ctx, loss);
}